// ProjectionAggregating_85839216378521
// MI455X (gfx1250) — compile-verified
//
#include <hip/hip_runtime.h>
#include <hip/hip_bf16.h>

typedef __attribute__((ext_vector_type(2))) float v2f;
typedef __attribute__((ext_vector_type(8))) float v8f;
typedef int v4i __attribute__((vector_size(4 * sizeof(int))));

#define BDIM  128
#define RNUM  64
#define LDSTR 132   // padded LDS row stride in floats (132 mod 64 = 4 banks/row)

#if __has_builtin(__builtin_amdgcn_global_load_async_to_lds_b128)
#define HAVE_ASYNC 1
#else
#define HAVE_ASYNC 0
#endif

#if HAVE_ASYNC
__device__ __forceinline__ void async_cp16(const float* g, float* l) {
    __builtin_amdgcn_global_load_async_to_lds_b128(
        (__attribute__((address_space(1))) v4i*)g,
        (__attribute__((address_space(3))) v4i*)l,
        /*offset=*/0, /*cpol=*/0);
}
template <int N>
__device__ __forceinline__ void wait_async() {
#if __has_builtin(__builtin_amdgcn_s_wait_asynccnt)
    __builtin_amdgcn_s_wait_asynccnt(N);
#else
    asm volatile("s_wait_asynccnt %0" :: "i"(N) : "memory");
#endif
}
#endif

__global__ __launch_bounds__(256) void proj_agg_kernel(
    const float* __restrict__ query_emb,
    const float* __restrict__ offset_emb,
    const float* __restrict__ refer_embs,
    const float* __restrict__ query_r,
    const float* __restrict__ refer_r,
    const float* __restrict__ start_embs,
    const float* __restrict__ W1,   // (128, 256) row-major: [k][d | d+128]
    const float* __restrict__ b1,   // (128)
    const float* __restrict__ W2,   // (128, 128) row-major: [d][k]
    const float* __restrict__ b2,   // (128)
    float* __restrict__ out_q,      // (B, 128)
    float* __restrict__ out_off)    // (B, 128)
{
    __shared__ float rr[RNUM * LDSTR]; // refer_r, later overwritten with bias
    __shared__ float hb[RNUM * LDSTR]; // h = relu(...)
#if HAVE_ASYNC
    __shared__ float reb[RNUM * LDSTR]; // refer_embs (async-staged)
    __shared__ float seb[RNUM * LDSTR]; // start_embs (async-staged)
#endif

    const int b    = blockIdx.x;
    const int tid  = threadIdx.x;
    const int wave = tid >> 5;
    const int lane = tid & 31;
    const int nn   = lane & 15;     // N index within tile (B/C/D); M index for A
    const int hi   = lane >> 4;     // 0: K pair {0,1} / rows 0-7 ; 1: K pair {2,3} / rows 8-15
    const int ksub = hi * 2;
    const int n0   = wave * 16;     // this wave's N-tile origin

    const float* rrG = refer_r    + (size_t)b * RNUM * BDIM;
    const float* reG = refer_embs + (size_t)b * RNUM * BDIM;
    const float* seG = start_embs + (size_t)b * RNUM * BDIM;

    // ---- Phase A: stage refer_r -> LDS; start streaming bias tensors ----
#if HAVE_ASYNC
    // Async DMA all three streaming tensors into LDS; overlap with qv + GEMM1.
    #pragma unroll
    for (int i = 0; i < 8; ++i) {
        int g   = tid + 256 * i;        // float4 index, 0..2047 (coalesced)
        int row = g >> 5;
        int c4  = (g & 31) << 2;
        async_cp16(rrG + g * 4, &rr[row * LDSTR + c4]);
    }
    #pragma unroll
    for (int i = 0; i < 8; ++i) {
        int g   = tid + 256 * i;
        int row = g >> 5;
        int c4  = (g & 31) << 2;
        async_cp16(reG + g * 4, &reb[row * LDSTR + c4]);
        async_cp16(seG + g * 4, &seb[row * LDSTR + c4]);
    }
#else
    #pragma unroll
    for (int i = 0; i < 8; ++i) {
        int g   = tid + 256 * i;
        int row = g >> 5;
        int c4  = (g & 31) << 2;
        float4 v = ((const float4*)rrG)[g];
        float* dst = &rr[row * LDSTR + c4];
        dst[0] = v.x; dst[1] = v.y; dst[2] = v.z; dst[3] = v.w;
    }
    // Prefetch the bias tensors so their HBM fetch overlaps GEMM1.
    #pragma unroll
    for (int i = 0; i < 8; ++i) {
        int g = tid + 256 * i;
        __builtin_prefetch(reG + g * 4, 0, 0);
        __builtin_prefetch(seG + g * 4, 0, 0);
    }
#endif
    if (tid < BDIM) {
        out_off[(size_t)b * BDIM + tid] = offset_emb[(size_t)b * BDIM + tid];
    }

    // ---- Per-lane scalar: qv = b1[k] + dot(query_r[b], W1b[k]),  k = n0+nn ----
    float qvn;
    {
        const int k      = n0 + nn;
        const float* qr  = query_r + (size_t)b * BDIM;
        const float* w1b = W1 + (size_t)k * (2 * BDIM) + BDIM;
        float acc = b1[k];
        #pragma unroll 8
        for (int d = 0; d < BDIM; d += 4) {
            acc += qr[d + 0] * w1b[d + 0] + qr[d + 1] * w1b[d + 1]
                 + qr[d + 2] * w1b[d + 2] + qr[d + 3] * w1b[d + 3];
        }
        qvn = acc;
    }

#if HAVE_ASYNC
    wait_async<16>();  // first 8 async ops (refer_r -> rr) complete, in-order
#endif
    __syncthreads();

    // ---- GEMM1: h = relu( refer_r (64x128) @ W1a^T (128x128) + qv ) ----
    {
        v2f bf[32];
        #pragma unroll
        for (int kk = 0; kk < 32; ++kk) {
            // B[d][k]: lane nn supplies column n0+nn; K slice 4*kk + {ksub, ksub+1}
            bf[kk] = *(const v2f*)(W1 + (size_t)(n0 + nn) * (2 * BDIM) + 4 * kk + ksub);
        }
        #pragma unroll
        for (int rt = 0; rt < 4; ++rt) {
            const int r0 = rt * 16;
            v8f acc;
            #pragma unroll
            for (int j = 0; j < 8; ++j) acc[j] = qvn;
            #pragma unroll
            for (int kk = 0; kk < 32; ++kk) {
                v2f a = *(const v2f*)(&rr[(r0 + nn) * LDSTR + 4 * kk + ksub]);
                acc = __builtin_amdgcn_wmma_f32_16x16x4_f32(
                        false, a, false, bf[kk], (short)0, acc, false, false);
            }
            #pragma unroll
            for (int j = 0; j < 8; ++j) {
                float v = acc[j] > 0.0f ? acc[j] : 0.0f;      // relu
                int row = r0 + j + hi * 8;                    // D layout: VGPR j -> M = j (+8 for hi lanes)
                hb[row * LDSTR + n0 + nn] = v;
            }
        }
    }

#if HAVE_ASYNC
    wait_async<0>();   // refer_embs / start_embs staged
#endif
    __syncthreads();

    // ---- bias = refer_embs - start_embs - refer_r (in place over rr) ----
    {
        #pragma unroll
        for (int i = 0; i < 8; ++i) {
            int g   = tid + 256 * i;
            int row = g >> 5;
            int c4  = (g & 31) << 2;
            float* p = &rr[row * LDSTR + c4];
#if HAVE_ASYNC
            const float4 e = *(const float4*)(&reb[row * LDSTR + c4]);
            const float4 s = *(const float4*)(&seb[row * LDSTR + c4]);
#else
            const float4 e = ((const float4*)reG)[g];
            const float4 s = ((const float4*)seG)[g];
#endif
            p[0] = e.x - s.x - p[0];
            p[1] = e.y - s.y - p[1];
            p[2] = e.z - s.z - p[2];
            p[3] = e.w - s.w - p[3];
        }
    }

    __syncthreads();

    // ---- GEMM2: attn = h @ W2^T + b2 ; fused reduction sum_r attn*bias ----
    {
        v2f bf[32];
        #pragma unroll
        for (int kk = 0; kk < 32; ++kk) {
            bf[kk] = *(const v2f*)(W2 + (size_t)(n0 + nn) * BDIM + 4 * kk + ksub);
        }
        const float b2n = b2[n0 + nn];
        float stot = 0.0f;
        #pragma unroll
        for (int rt = 0; rt < 4; ++rt) {
            const int r0 = rt * 16;
            v8f acc;
            #pragma unroll
            for (int j = 0; j < 8; ++j) acc[j] = b2n;
            #pragma unroll
            for (int kk = 0; kk < 32; ++kk) {
                v2f a = *(const v2f*)(&hb[(r0 + nn) * LDSTR + 4 * kk + ksub]);
                acc = __builtin_amdgcn_wmma_f32_16x16x4_f32(
                        false, a, false, bf[kk], (short)0, acc, false, false);
            }
            #pragma unroll
            for (int j = 0; j < 8; ++j) {
                int row = r0 + j + hi * 8;
                stot += acc[j] * rr[row * LDSTR + n0 + nn];   // attn * bias
            }
        }
        // lanes L and L^16 hold partials for the same output column d = n0+nn
        stot += __shfl_xor(stot, 16, 32);
        if (lane < 16) {
            size_t idx = (size_t)b * BDIM + n0 + lane;
            out_q[idx] = query_emb[idx] + stot;
        }
    }
}

extern "C" void kernel_launch(void* const* d_in, const int* in_sizes, int n_in,
                              void* d_out, int out_size, void* d_ws, size_t ws_size,
                              hipStream_t stream) {
    const float* query_emb  = (const float*)d_in[0];
    const float* offset_emb = (const float*)d_in[1];
    const float* refer_embs = (const float*)d_in[2];
    const float* query_r    = (const float*)d_in[3];
    const float* refer_r    = (const float*)d_in[4];
    const float* start_embs = (const float*)d_in[5];
    const float* W1         = (const float*)d_in[6];
    const float* b1         = (const float*)d_in[7];
    const float* W2         = (const float*)d_in[8];
    const float* b2         = (const float*)d_in[9];

    float* out = (float*)d_out;
    const int Bn = in_sizes[0] / BDIM;           // 4096
    float* out_q   = out;
    float* out_off = out + (size_t)Bn * BDIM;    // second tuple element, flat-concatenated

    proj_agg_kernel<<<Bn, 256, 0, stream>>>(
        query_emb, offset_emb, refer_embs, query_r, refer_r, start_embs,
        W1, b1, W2, b2, out_q, out_off);
}